// MPNN_89756226552533
// MI455X (gfx1250) — compile-verified
//
#include <hip/hip_runtime.h>
#include <hip/hip_bf16.h>

#define NWAVE   8
#define NCH     18      // 2*(NWAVE+1)
#define CST     20      // padded iter_coeff row stride (80B, float4-aligned)
#define NORB    24      // NWAVE*(MAX_L+1)
#define RH      64
#define C00  0.28209479177387814f
#define C1   0.4886025119029199f
#define C2A  1.0925484305920792f
#define C2B  0.31539156525252005f
#define C2C  0.5462742152960396f

typedef __attribute__((ext_vector_type(2))) float v2f;
typedef __attribute__((ext_vector_type(8))) float v8f;

__device__ __forceinline__ v8f wmma4(v2f a, v2f b, v8f c) {
    // D = A(16x4) * B(4x16) + C(16x16), all f32
    return __builtin_amdgcn_wmma_f32_16x16x4_f32(false, a, false, b, (short)0, c, false, false);
}

#if __has_builtin(__builtin_amdgcn_tanhf)
__device__ __forceinline__ float fast_tanh(float x) { return __builtin_amdgcn_tanhf(x); }
#else
__device__ __forceinline__ float fast_tanh(float x) { return tanhf(x); }
#endif

#if __has_builtin(__builtin_amdgcn_cosf)
// hw cos computes cos(2*pi*x); we need cos(d*pi/5) = cos_hw(d * 0.1)
__device__ __forceinline__ float cut_cos(float dist) { return __builtin_amdgcn_cosf(dist * 0.1f); }
#else
__device__ __forceinline__ float cut_cos(float dist) { return cosf(dist * 0.6283185307179586f); }
#endif

// ---------------------------------------------------------------------------
// iter_coeff[i][c] = emb_table[species[i]][c]  (rows padded to CST floats)
__global__ void mpnn_init_coeff(const int* __restrict__ species,
                                const float* __restrict__ emb,
                                float* __restrict__ coeff, int n) {
    int i = blockIdx.x * blockDim.x + threadIdx.x;
    if (i >= n * CST) return;
    int node = i / CST, c = i - node * CST;
    coeff[i] = (c < NCH) ? emb[species[node] * NCH + c] : 0.0f;
}

// out[0] = n * bo  (also clears the poisoned output)
__global__ void mpnn_out_init(float* __restrict__ out, const float* __restrict__ bo, int n) {
    if (threadIdx.x == 0 && blockIdx.x == 0) out[0] = (float)n * bo[0];
}

// ---------------------------------------------------------------------------
// Per-edge: distvec = cart[idx_n]-cart[idx_c]+shifts ; cut = (0.5cos(d*pi/5)+0.5)^3
__global__ void mpnn_edge_pre(const float* __restrict__ cart,
                              const float* __restrict__ shifts,
                              const int* __restrict__ idx_c,
                              const int* __restrict__ idx_n,
                              float4* __restrict__ edge4, int nEdges) {
    int e = blockIdx.x * blockDim.x + threadIdx.x;
    if (e >= nEdges) return;
    int c = idx_c[e], nn = idx_n[e];
    float dx = cart[3*nn+0] - cart[3*c+0] + shifts[3*e+0];
    float dy = cart[3*nn+1] - cart[3*c+1] + shifts[3*e+1];
    float dz = cart[3*nn+2] - cart[3*c+2] + shifts[3*e+2];
    float dist = sqrtf(dx*dx + dy*dy + dz*dz);
    float t = 0.5f * cut_cos(dist) + 0.5f;
    float cut = t * t * t;
    edge4[e] = make_float4(dx, dy, dz, cut);
}

// ---------------------------------------------------------------------------
// Edge scatter: 11 b128 gathers + 32 L2-resident f32 atomics per edge.
__global__ void mpnn_edge_scatter(const float* __restrict__ coeff,
                                  const float* __restrict__ mp_prev,
                                  const float4* __restrict__ edge4,
                                  const int* __restrict__ idx_c,
                                  const int* __restrict__ idx_n,
                                  float* __restrict__ mp_dis,
                                  float* __restrict__ mp_new, int nEdges) {
    int e = blockIdx.x * blockDim.x + threadIdx.x;
    if (e >= nEdges) return;
    int c = idx_c[e], nn = idx_n[e];
    float4 ed = edge4[e];
    float cut = ed.w;

    const float4* ncv = (const float4*)(coeff + (size_t)nn * CST);
    float4 q0 = ncv[0], q1 = ncv[1], q2 = ncv[2], q3 = ncv[3], q4 = ncv[4];
    const float4* mpv = (const float4*)(mp_prev + (size_t)nn * NORB);
    float4 p0 = mpv[0], p1 = mpv[1], p2 = mpv[2], p3 = mpv[3], p4 = mpv[4], p5 = mpv[5];

    float* dis = mp_dis + (size_t)c * NWAVE;
    float* mpc = mp_new + (size_t)c * NORB;

    // wd = cut * nc[0:8]
    atomicAdd(&dis[0], cut * q0.x); atomicAdd(&dis[1], cut * q0.y);
    atomicAdd(&dis[2], cut * q0.z); atomicAdd(&dis[3], cut * q0.w);
    atomicAdd(&dis[4], cut * q1.x); atomicAdd(&dis[5], cut * q1.y);
    atomicAdd(&dis[6], cut * q1.z); atomicAdd(&dis[7], cut * q1.w);

    // w[k] = cut * nc[9+k]
    float w[8]  = { cut*q2.y, cut*q2.z, cut*q2.w, cut*q3.x,
                    cut*q3.y, cut*q3.z, cut*q3.w, cut*q4.x };
    float px[8] = { p0.x, p0.y, p0.z, p0.w, p1.x, p1.y, p1.z, p1.w };
    float py[8] = { p2.x, p2.y, p2.z, p2.w, p3.x, p3.y, p3.z, p3.w };
    float pz[8] = { p4.x, p4.y, p4.z, p4.w, p5.x, p5.y, p5.z, p5.w };
#pragma unroll
    for (int k = 0; k < NWAVE; ++k) {
        atomicAdd(&mpc[k],      ed.x * w[k] + px[k]);
        atomicAdd(&mpc[8 + k],  ed.y * w[k] + py[k]);
        atomicAdd(&mpc[16 + k], ed.z * w[k] + pz[k]);
    }
}

// ---------------------------------------------------------------------------
// Node phase: 8 waves/block, 16 nodes/wave. WMMA f32 16x16x4 for both MLP
// GEMMs and the Wm GEMM. H (16x64) staged via LDS between GEMMs.
__launch_bounds__(256)
__global__ void mpnn_node_update(int step,
                                 float* __restrict__ coeff,
                                 const float* __restrict__ mp_dis,
                                 const float* __restrict__ mp_new,
                                 float* __restrict__ density,
                                 const float* __restrict__ rw1, const float* __restrict__ rb1,
                                 const float* __restrict__ rw2, const float* __restrict__ rb2,
                                 const float* __restrict__ Wm,  const float* __restrict__ bm,
                                 const float* __restrict__ Wo,
                                 float* __restrict__ out, int n) {
    __shared__ float sRw1[8 * 64];       // rw1 row-major
    __shared__ float sRw2[64 * 16];      // rw2 padded N 8->16
    __shared__ float sWm[24 * 32];       // Wm[step] padded N 18->32
    __shared__ float sH[8][16 * 68];     // per-wave: H (16x64), later density (16x24)
    __shared__ float sRed[256];

    const int tid = threadIdx.x;
    for (int i = tid; i < 8 * 64; i += 256) sRw1[i] = rw1[i];
    for (int i = tid; i < 64 * 16; i += 256) {
        int r = i >> 4, cc = i & 15;
        sRw2[i] = (cc < NWAVE) ? rw2[r * NWAVE + cc] : 0.0f;
    }
    for (int i = tid; i < 24 * 32; i += 256) {
        int r = i >> 5, cc = i & 31;
        sWm[i] = (step < 2 && cc < NCH) ? Wm[step * NORB * NCH + r * NCH + cc] : 0.0f;
    }
    __syncthreads();

    const int wave = tid >> 5;
    const int lane = tid & 31;
    const int row  = lane & 15;          // A-row / D-column index
    const int col  = row;                // same value, used as N index
    const int kb   = (lane >> 4) * 2;    // K-pair select within 4-wide K group
    const int mh   = (lane >> 4) * 8;    // D-row offset for this half-wave
    const int nodeBase = blockIdx.x * 128 + wave * 16;

    // ---- GEMM1: H = tanh(MP_cd @ rw1 + rb1), MP_cd = mp_dis * coeff[:,8]
    int nodeA = nodeBase + row;
    int ndA = (nodeA < n) ? nodeA : (n - 1);
    bool vA = nodeA < n;
    float c8 = coeff[(size_t)ndA * CST + NWAVE];
    const float* disRow = mp_dis + (size_t)ndA * NWAVE;
    v2f a0, a1;
    a0.x = vA ? disRow[kb + 0] * c8 : 0.0f;
    a0.y = vA ? disRow[kb + 1] * c8 : 0.0f;
    a1.x = vA ? disRow[4 + kb + 0] * c8 : 0.0f;
    a1.y = vA ? disRow[4 + kb + 1] * c8 : 0.0f;

#pragma unroll
    for (int t = 0; t < 4; ++t) {
        v2f b0, b1;
        b0.x = sRw1[(kb + 0) * 64 + t * 16 + col];
        b0.y = sRw1[(kb + 1) * 64 + t * 16 + col];
        b1.x = sRw1[(4 + kb + 0) * 64 + t * 16 + col];
        b1.y = sRw1[(4 + kb + 1) * 64 + t * 16 + col];
        v8f acc = {};
        acc = wmma4(a0, b0, acc);
        acc = wmma4(a1, b1, acc);
        float bias = rb1[t * 16 + col];
#pragma unroll
        for (int i = 0; i < 8; ++i)
            sH[wave][(i + mh) * 68 + t * 16 + col] = fast_tanh(acc[i] + bias);
    }
    __syncthreads();

    // ---- GEMM2: R = H @ rw2 (K=64 -> 16 k-steps, 2 independent chains for ILP)
    v8f rA = {}, rB = {};
#pragma unroll
    for (int s = 0; s < 8; ++s) {
        v2f aa, bb, cc, dd;
        aa.x = sH[wave][row * 68 + s * 4 + kb + 0];
        aa.y = sH[wave][row * 68 + s * 4 + kb + 1];
        bb.x = sRw2[(s * 4 + kb + 0) * 16 + col];
        bb.y = sRw2[(s * 4 + kb + 1) * 16 + col];
        cc.x = sH[wave][row * 68 + (s + 8) * 4 + kb + 0];
        cc.y = sH[wave][row * 68 + (s + 8) * 4 + kb + 1];
        dd.x = sRw2[((s + 8) * 4 + kb + 0) * 16 + col];
        dd.y = sRw2[((s + 8) * 4 + kb + 1) * 16 + col];
        rA = wmma4(aa, bb, rA);
        rB = wmma4(cc, dd, rB);
    }
    v8f r = rA + rB;

    // ---- angular power + density update + staging for Wm GEMM
    float partial = 0.0f;
    if (col < NWAVE) {
        float wo0 = Wo[col], wo1 = Wo[8 + col], wo2 = Wo[16 + col];
        float rb2c = rb2[col];
#pragma unroll
        for (int i = 0; i < 8; ++i) {
            int m  = i + mh;
            int nd = nodeBase + m;
            bool v = nd < n;
            int ndc = v ? nd : (n - 1);
            const float* mp = mp_new + (size_t)ndc * NORB;
            float c17 = coeff[(size_t)ndc * CST + (NCH - 1)];
            float x = mp[col] * c17, y = mp[8 + col] * c17, z = mp[16 + col] * c17;
            float r2 = x * x + y * y + z * z;
            float a0c = C00 * C00;
            float a1c = C1 * C1 * r2;
            float t1 = C2A * x * y, t2 = C2A * y * z, t3 = C2A * x * z;
            float t4 = C2B * (3.0f * z * z - r2), t5 = C2C * (x * x - y * y);
            float a2c = t1 * t1 + t2 * t2 + t3 * t3 + t4 * t4 + t5 * t5;
            float rad = r[i] + rb2c;
            float* drow = density + (size_t)ndc * NORB;
            float d0 = drow[col]      + rad * a0c;
            float d1 = drow[8 + col]  + rad * a1c;
            float d2 = drow[16 + col] + rad * a2c;
            if (v) { drow[col] = d0; drow[8 + col] = d1; drow[16 + col] = d2; }
            sH[wave][m * 68 + col]      = d0;
            sH[wave][m * 68 + 8 + col]  = d1;
            sH[wave][m * 68 + 16 + col] = d2;
            if (step == 2 && v) partial += d0 * wo0 + d1 * wo1 + d2 * wo2;
        }
    }

    if (step < 2) {
        __syncthreads();
        // iter_coeff = tanh(density @ Wm[step] + bm[step]); K=24 -> 6 k-steps, N padded 18->32
        v8f ic0 = {}, ic1 = {};
#pragma unroll
        for (int s = 0; s < 6; ++s) {
            v2f aa, b0, b1;
            aa.x = sH[wave][row * 68 + s * 4 + kb + 0];
            aa.y = sH[wave][row * 68 + s * 4 + kb + 1];
            b0.x = sWm[(s * 4 + kb + 0) * 32 + col];
            b0.y = sWm[(s * 4 + kb + 1) * 32 + col];
            b1.x = sWm[(s * 4 + kb + 0) * 32 + 16 + col];
            b1.y = sWm[(s * 4 + kb + 1) * 32 + 16 + col];
            ic0 = wmma4(aa, b0, ic0);
            ic1 = wmma4(aa, b1, ic1);
        }
#pragma unroll
        for (int i = 0; i < 8; ++i) {
            int nd = nodeBase + i + mh;
            if (nd < n) {
                coeff[(size_t)nd * CST + col] = fast_tanh(ic0[i] + bm[step * NCH + col]);
                int c2 = 16 + col;
                if (c2 < NCH)
                    coeff[(size_t)nd * CST + c2] = fast_tanh(ic1[i] + bm[step * NCH + c2]);
            }
        }
    } else {
        // final scalar reduction
        sRed[tid] = partial;
        __syncthreads();
        for (int off = 128; off > 0; off >>= 1) {
            if (tid < off) sRed[tid] += sRed[tid + off];
            __syncthreads();
        }
        if (tid == 0) atomicAdd(out, sRed[0]);
    }
}

// ---------------------------------------------------------------------------
extern "C" void kernel_launch(void* const* d_in, const int* in_sizes, int n_in,
                              void* d_out, int out_size, void* d_ws, size_t ws_size,
                              hipStream_t stream) {
    const float* cart   = (const float*)d_in[0];
    const float* shifts = (const float*)d_in[1];
    const float* emb    = (const float*)d_in[2];
    const float* rw1    = (const float*)d_in[3];
    const float* rb1    = (const float*)d_in[4];
    const float* rw2    = (const float*)d_in[5];
    const float* rb2    = (const float*)d_in[6];
    const float* Wm     = (const float*)d_in[7];
    const float* bm     = (const float*)d_in[8];
    const float* Wo     = (const float*)d_in[9];
    const float* bo     = (const float*)d_in[10];
    const int*   neigh  = (const int*)d_in[11];
    const int*   spec   = (const int*)d_in[12];
    float*       out    = (float*)d_out;

    const int n = in_sizes[12];           // N_ATOMS
    const int E = in_sizes[11] / 2;       // N_EDGES
    const int* idx_c = neigh;
    const int* idx_n = neigh + E;

    size_t off = 0;
    auto wsalloc = [&](size_t bytes) {
        void* p = (char*)d_ws + off;
        off += (bytes + 255) & ~(size_t)255;
        return p;
    };
    float4* edge4 = (float4*)wsalloc(sizeof(float) * 4 * (size_t)E);
    float*  coeff = (float*)wsalloc(sizeof(float) * CST  * (size_t)n);
    float*  mpdis = (float*)wsalloc(sizeof(float) * NWAVE * (size_t)n);
    float*  mpA   = (float*)wsalloc(sizeof(float) * NORB * (size_t)n);  // MP_cart prev
    float*  mpB   = (float*)wsalloc(sizeof(float) * NORB * (size_t)n);  // MP_cart new
    float*  dens  = (float*)wsalloc(sizeof(float) * NORB * (size_t)n);

    hipMemsetAsync(mpA,  0, sizeof(float) * NORB * (size_t)n, stream);
    hipMemsetAsync(dens, 0, sizeof(float) * NORB * (size_t)n, stream);

    mpnn_init_coeff<<<(n * CST + 255) / 256, 256, 0, stream>>>(spec, emb, coeff, n);
    mpnn_edge_pre<<<(E + 255) / 256, 256, 0, stream>>>(cart, shifts, idx_c, idx_n, edge4, E);
    mpnn_out_init<<<1, 32, 0, stream>>>(out, bo, n);

    float* mpPrev = mpA;
    float* mpNew  = mpB;
    const int nodeBlocks = (n + 127) / 128;
    for (int step = 0; step < 3; ++step) {
        hipMemsetAsync(mpdis, 0, sizeof(float) * NWAVE * (size_t)n, stream);
        hipMemsetAsync(mpNew, 0, sizeof(float) * NORB  * (size_t)n, stream);
        mpnn_edge_scatter<<<(E + 255) / 256, 256, 0, stream>>>(
            coeff, mpPrev, edge4, idx_c, idx_n, mpdis, mpNew, E);
        mpnn_node_update<<<nodeBlocks, 256, 0, stream>>>(
            step, coeff, mpdis, mpNew, dens, rw1, rb1, rw2, rb2, Wm, bm, Wo, out, n);
        float* tmp = mpPrev; mpPrev = mpNew; mpNew = tmp;
    }
}